// FP8SparseLinearLayer_78486232367565
// MI455X (gfx1250) — compile-verified
//
#include <hip/hip_runtime.h>
#include <stdint.h>

typedef _Float16 half16 __attribute__((ext_vector_type(16)));
typedef _Float16 half32 __attribute__((ext_vector_type(32)));
typedef float    floatx8 __attribute__((ext_vector_type(8)));
typedef unsigned int uintx4 __attribute__((ext_vector_type(4)));
typedef int      intx4 __attribute__((ext_vector_type(4)));
typedef int      intx8 __attribute__((ext_vector_type(8)));

#define K_DIM   1024
#define O_DIM   1024
#define T_ROWS  65536
#define T_BLK   64
#define KCHUNK  256
#define GRP_STRIDE 260            // dwords per kk-quad group (1040 B, 16B-aligned, padded)
#define N_GROUPS   128            // 512 kk rows / 4
#define WC_HALF_PER_ROW 512       // K/2 compressed halves per output row
#define IDX_WORDS_PER_ROW 32      // K/32 index words per output row
#define WC_BYTES (O_DIM * WC_HALF_PER_ROW * 2)   // 1 MB

static __device__ inline unsigned pack_f16x2(float a, float b) {
    union { _Float16 h[2]; unsigned u; } p;
    p.h[0] = (_Float16)a; p.h[1] = (_Float16)b;
    return p.u;
}

// ---------------------------------------------------------------------------
// Prep: 2:4 soft-threshold + scale -> compressed f16 values + 2-bit indices.
// One thread handles 32 K (8 groups of 4) of one output row and emits one
// GEMM-ready 8-nibble index word (matches SWMMAC lane index layout).
// ---------------------------------------------------------------------------
__global__ void __launch_bounds__(256)
prep24_kernel(const float* __restrict__ w, const float* __restrict__ scale_p,
              _Float16* __restrict__ wc, unsigned int* __restrict__ idx) {
    const float s = scale_p[0];
    const int gid = blockIdx.x * blockDim.x + threadIdx.x;  // O_DIM * 32 threads
    const int o  = gid >> 5;
    const int wi = gid & 31;
    const float4* wp = (const float4*)(w + (size_t)o * K_DIM + wi * 32);
    unsigned int code_word = 0;
    union { _Float16 h[16]; uintx4 q[2]; } vals;
#pragma unroll
    for (int g = 0; g < 8; ++g) {
        float4 v = wp[g];
        float a0 = fabsf(v.x), a1 = fabsf(v.y), a2 = fabsf(v.z), a3 = fabsf(v.w);
        float lo01 = fminf(a0, a1), hi01 = fmaxf(a0, a1);
        float lo23 = fminf(a2, a3), hi23 = fmaxf(a2, a3);
        // 2nd smallest == 3rd largest magnitude
        float t = fminf(fmaxf(lo01, lo23), fminf(hi01, hi23));
        // positions of the two largest magnitudes (stable)
        int i0 = 0; float m0 = a0;
        if (a1 > m0) { m0 = a1; i0 = 1; }
        if (a2 > m0) { m0 = a2; i0 = 2; }
        if (a3 > m0) { m0 = a3; i0 = 3; }
        int i1 = -1; float m1 = -1.0f;
        if (i0 != 0 && a0 > m1) { m1 = a0; i1 = 0; }
        if (i0 != 1 && a1 > m1) { m1 = a1; i1 = 1; }
        if (i0 != 2 && a2 > m1) { m1 = a2; i1 = 2; }
        if (i0 != 3 && a3 > m1) { m1 = a3; i1 = 3; }
        int lo = min(i0, i1), hi = max(i0, i1);
        float av[4] = { v.x, v.y, v.z, v.w };
        float aa[4] = { a0, a1, a2, a3 };
        float v0 = copysignf(fmaxf(aa[lo] - t, 0.0f), av[lo]) * s;
        float v1 = copysignf(fmaxf(aa[hi] - t, 0.0f), av[hi]) * s;
        vals.h[2 * g]     = (_Float16)v0;
        vals.h[2 * g + 1] = (_Float16)v1;
        code_word |= (unsigned)(lo | (hi << 2)) << (4 * g);
    }
    uintx4* wq = (uintx4*)(wc + (size_t)o * WC_HALF_PER_ROW + wi * 16);
    wq[0] = vals.q[0];
    wq[1] = vals.q[1];
    idx[o * IDX_WORDS_PER_ROW + wi] = code_word;
}

// ---------------------------------------------------------------------------
// SWMMAC GEMM: grid = T_ROWS/T_BLK blocks; 256 threads (8 waves).
// Each block: TDM-stage 64 x-rows (full K) to LDS as packed f16 pairs,
// then loop all O in 128-row slices (wave -> 16 O x 64 T via 4 accumulators).
// B tiles are loaded 2-3 tiles ahead of their consuming SWMMAC so their live
// ranges overlap -> distinct registers -> partial dscnt waits (latency hidden).
// ---------------------------------------------------------------------------
__global__ void __launch_bounds__(256)
sparse_swmmac_gemm(const float* __restrict__ x, const _Float16* __restrict__ wc,
                   const unsigned int* __restrict__ idx,
                   const float* __restrict__ bias, float* __restrict__ out) {
    extern __shared__ char smem[];
    float* stag = (float*)smem;                                   // 64 KB fp32 staging
    unsigned int* xb = (unsigned int*)(smem + KCHUNK * T_BLK * 4); // group-of-4 layout

    const int tid  = threadIdx.x;
    const int wave = tid >> 5;
    const int lane = tid & 31;
    const int r = lane & 15;   // row within 16 (A) / column N (B,C)
    const int h = lane >> 4;   // half-wave selector
    const long t_base = (long)blockIdx.x * T_BLK;

    // ---- Stage x tile: TDM fp32 chunk -> LDS, then convert to packed f16 ----
    for (int c = 0; c < K_DIM / KCHUNK; ++c) {
        if (tid == 0) {
            unsigned long long ga =
                (unsigned long long)(const void*)(x + t_base * K_DIM + c * KCHUNK);
            unsigned lds_off = (unsigned)(size_t)stag;
            uintx4 g0;
            g0[0] = 1u;                                   // count=1, user descriptor
            g0[1] = lds_off;                              // lds_addr (bytes)
            g0[2] = (unsigned)(ga & 0xffffffffu);         // global_addr lo
            g0[3] = (unsigned)((ga >> 32) & 0x01ffffffu) | (2u << 30); // addr hi | type=2
            intx8 g1;
            g1[0] = 0x00020000;                 // data_size=4B; no pad/iterate/multicast
            g1[1] = (int)(K_DIM << 16);         // tensor_dim0 = 1024 (low 16 bits)
            g1[2] = 0;                          // tensor_dim0 hi, tensor_dim1 lo16
            g1[3] = (int)((KCHUNK << 16) | 1);  // tile_dim0=256; tensor_dim1=65536 hi16
            g1[4] = T_BLK;                      // tile_dim1 = 64 rows
            g1[5] = K_DIM;                      // tensor_dim0_stride = 1024
            g1[6] = 0;
            g1[7] = 0;
            intx4 z4 = { 0, 0, 0, 0 };
            intx8 z8 = { 0, 0, 0, 0, 0, 0, 0, 0 };
            __builtin_amdgcn_tensor_load_to_lds(g0, g1, z4, z4, z8, 0);
            __builtin_amdgcn_s_wait_tensorcnt(0);
        }
        __syncthreads();
        // fp32 -> packed f16 pairs: xb[group][t][kk&3], group = kk>>2
        {
            const int kkl = tid & 63;
            const int t0  = tid >> 6;  // 0..3
            const int gq  = c * 32 + (kkl >> 2);
            const int rr  = kkl & 3;
#pragma unroll
            for (int p = 0; p < 16; ++p) {
                int t = t0 + p * 4;
                float2 f0 = ((const float2*)stag)[t * 128 + kkl];
                float2 f1 = ((const float2*)stag)[t * 128 + 64 + kkl];
                xb[gq * GRP_STRIDE + t * 4 + rr]        = pack_f16x2(f0.x, f0.y);
                xb[(gq + 16) * GRP_STRIDE + t * 4 + rr] = pack_f16x2(f1.x, f1.y);
            }
        }
        __syncthreads();
    }

    // ---- Main loop over output channels: 8 slices of 128 O-rows ----
    for (int i = 0; i < O_DIM / 128; ++i) {
        const int o_w = i * 128 + wave * 16;
        floatx8 acc[4] = { floatx8{}, floatx8{}, floatx8{}, floatx8{} };
        union { half16 v; uintx4 q[2]; } au;
        union BU { half32 v; uintx4 q4[4]; } bt[4];

#pragma unroll 4
        for (int k64 = 0; k64 < K_DIM; k64 += 64) {
            // A: compressed 16x32 halves of the 16x64 sparse tile (L2-resident)
            const _Float16* arow =
                wc + (size_t)(o_w + r) * WC_HALF_PER_ROW + (k64 >> 1) + h * 8;
            au.q[0] = *(const uintx4*)(arow);
            au.q[1] = *(const uintx4*)(arow + 16);
            const int iw =
                (int)idx[(o_w + r) * IDX_WORDS_PER_ROW + (k64 >> 5) + h];
            const int gbase = (k64 >> 3) + h * 2;   // (k64/2)/4 quads

            auto loadB = [&](BU& b, int nt) {
                const int cb4 = (nt * 16 + r) * 4;
                b.q4[0] = *(const uintx4*)&xb[(gbase    ) * GRP_STRIDE + cb4];
                b.q4[1] = *(const uintx4*)&xb[(gbase + 1) * GRP_STRIDE + cb4];
                b.q4[2] = *(const uintx4*)&xb[(gbase + 4) * GRP_STRIDE + cb4];
                b.q4[3] = *(const uintx4*)&xb[(gbase + 5) * GRP_STRIDE + cb4];
            };

            // Software pipeline: keep 2-3 B tiles in flight past each SWMMAC.
            loadB(bt[0], 0);
            loadB(bt[1], 1);
            loadB(bt[2], 2);
            acc[0] = __builtin_amdgcn_swmmac_f32_16x16x64_f16(
                false, au.v, false, bt[0].v, acc[0], iw, false, false);
            loadB(bt[3], 3);
            acc[1] = __builtin_amdgcn_swmmac_f32_16x16x64_f16(
                false, au.v, false, bt[1].v, acc[1], iw, false, false);
            acc[2] = __builtin_amdgcn_swmmac_f32_16x16x64_f16(
                false, au.v, false, bt[2].v, acc[2], iw, false, false);
            acc[3] = __builtin_amdgcn_swmmac_f32_16x16x64_f16(
                false, au.v, false, bt[3].v, acc[3], iw, false, false);
        }

        // ---- Epilogue: transpose via LDS, fuse bias, coalesced stores ----
        __syncthreads();
        float* trans = (float*)smem;  // 64 x 132 fp32 (reuses staging region)
#pragma unroll
        for (int nt = 0; nt < 4; ++nt) {
#pragma unroll
            for (int v = 0; v < 8; ++v) {
                int ol = wave * 16 + v + h * 8;
                int tl = nt * 16 + r;
                trans[tl * 132 + ol] = acc[nt][v];
            }
        }
        __syncthreads();
        {
            const int ol = tid & 127;
            const int th = tid >> 7;
            const float bo = bias[i * 128 + ol];
#pragma unroll
            for (int t = th; t < T_BLK; t += 2) {
                out[(t_base + t) * O_DIM + i * 128 + ol] = trans[t * 132 + ol] + bo;
            }
        }
        __syncthreads();
    }
}

// ---------------------------------------------------------------------------
extern "C" void kernel_launch(void* const* d_in, const int* in_sizes, int n_in,
                              void* d_out, int out_size, void* d_ws, size_t ws_size,
                              hipStream_t stream) {
    const float* x     = (const float*)d_in[0];
    const float* w     = (const float*)d_in[1];
    const float* bias  = (const float*)d_in[2];
    const float* scale = (const float*)d_in[3];
    float* out = (float*)d_out;

    _Float16* wc = (_Float16*)d_ws;
    unsigned int* idx = (unsigned int*)((char*)d_ws + WC_BYTES);

    prep24_kernel<<<dim3((O_DIM * 32) / 256), dim3(256), 0, stream>>>(w, scale, wc, idx);

    size_t smem_bytes = (size_t)KCHUNK * T_BLK * 4 + (size_t)N_GROUPS * GRP_STRIDE * 4;
    sparse_swmmac_gemm<<<dim3(T_ROWS / T_BLK), dim3(256), smem_bytes, stream>>>(
        x, wc, idx, bias, out);
}